// SparseLinearLayer_2903397892397
// MI455X (gfx1250) — compile-verified
//
#include <hip/hip_runtime.h>
#include <hip/hip_bf16.h>

typedef float v2f __attribute__((ext_vector_type(2)));
typedef float v8f __attribute__((ext_vector_type(8)));

#define KEEP_PROB 0.9f
#define OUT_DIM   128
#define TILE_ROWS 16

__device__ __forceinline__ int lower_bound_i32(const int* __restrict__ a, int n, int target) {
    int lo = 0, hi = n;
    while (lo < hi) {
        int mid = (lo + hi) >> 1;
        if (a[mid] < target) lo = mid + 1; else hi = mid;
    }
    return lo;
}

// Build one K=4 A-fragment (+ the two weight-row indices this lane needs).
// GUARD=false: no bounds checks at all (hot path, branch-free).
// GUARD=true : clamped indices + cndmask selects (still branch-free).
template <bool GUARD>
__device__ __forceinline__ void load_chunk(const float* __restrict__ values,
                                           const int*   __restrict__ rows,
                                           const int*   __restrict__ cols,
                                           const unsigned char* __restrict__ keep_mask,
                                           int k, int kEnd, int r0, int m, int half,
                                           float inv_keep,
                                           v2f& a, int& c0, int& c1)
{
    // A layout: half0 holds K=0,1 ; half1 holds K=2,3.
    // B layout (VGPR0=K{0,2}, VGPR1=K{1,3}) uses the same k offsets.
    const int kk0 = k + 2 * half;
    const int kk1 = kk0 + 1;

    int  i0 = kk0, i1 = kk1;
    bool ok0 = true, ok1 = true;
    if (GUARD) {
        const int last = kEnd - 1;          // kEnd > kStart >= 0 here
        ok0 = kk0 < kEnd;
        ok1 = kk1 < kEnd;
        i0  = ok0 ? kk0 : last;             // clamped -> loads always in-bounds
        i1  = ok1 ? kk1 : last;
    }

    // Unconditional loads; dropout + segment masking via selects only.
    const int   rr0 = rows[i0];
    const int   rr1 = rows[i1];
    c0 = cols[i0];
    c1 = cols[i1];
    float v0 = values[i0] * inv_keep;
    float v1 = values[i1] * inv_keep;
    v0 = keep_mask[i0] ? v0 : 0.f;
    v1 = keep_mask[i1] ? v1 : 0.f;
    if (GUARD) {
        v0 = ok0 ? v0 : 0.f;
        v1 = ok1 ? v1 : 0.f;
    }
    a.x = (rr0 == r0 + m) ? v0 : 0.f;       // one-hot scatter row of A
    a.y = (rr1 == r0 + m) ? v1 : 0.f;
}

// Gather B fragments for all 8 column tiles and accumulate 8 WMMAs.
__device__ __forceinline__ void mma_chunk(const float* __restrict__ weights,
                                          int m, const v2f& a, int c0, int c1,
                                          v8f acc[8])
{
    const float* w0 = weights + (size_t)c0 * OUT_DIM + m;
    const float* w1 = weights + (size_t)c1 * OUT_DIM + m;
    v2f b[8];
#pragma unroll
    for (int nt = 0; nt < 8; ++nt) {
        b[nt].x = w0[nt * 16];              // B[K = 2*half    , n = m]
        b[nt].y = w1[nt * 16];              // B[K = 2*half + 1, n = m]
    }
#pragma unroll
    for (int nt = 0; nt < 8; ++nt) {
        acc[nt] = __builtin_amdgcn_wmma_f32_16x16x4_f32(
            false, a, false, b[nt], (short)0, acc[nt], false, false);
    }
}

// One wave32 per 16-row output tile. Rows are sorted, so each tile owns a
// contiguous nnz segment -> no atomics. out_tile(16x128) = A(16xK) x B(Kx128)
// accumulated K=4 at a time with V_WMMA_F32_16X16X4_F32 (exact fp32).
__global__ __launch_bounds__(32)
void spmm_wmma_f32_kernel(const float* __restrict__ values,
                          const int*   __restrict__ rows,
                          const int*   __restrict__ cols,
                          const unsigned char* __restrict__ keep_mask,
                          const float* __restrict__ weights,
                          const float* __restrict__ bias,
                          float* __restrict__ out,
                          int nnz, int n_nodes)
{
    const int tile = blockIdx.x;
    const int r0   = tile * TILE_ROWS;
    if (r0 >= n_nodes) return;

    const int lane = threadIdx.x & 31;
    const int half = lane >> 4;   // 0: lanes 0-15, 1: lanes 16-31
    const int m    = lane & 15;   // A: row-in-tile; B/C: col-in-16-tile

    const int kStart = lower_bound_i32(rows, nnz, r0);
    const int kEnd   = lower_bound_i32(rows, nnz, r0 + TILE_ROWS);

    const v8f vzero = {0.f, 0.f, 0.f, 0.f, 0.f, 0.f, 0.f, 0.f};
    v8f acc[8];
#pragma unroll
    for (int nt = 0; nt < 8; ++nt) acc[nt] = vzero;

    const float inv_keep = 1.0f / KEEP_PROB;
    const int   kFull    = kStart + ((kEnd - kStart) & ~3);   // full K=4 chunks

    int k = kStart;

    // Hot loop: K=8 per iteration, completely branch/guard free.
    for (; k + 8 <= kEnd; k += 8) {
        v2f aA, aB;
        int c0A, c1A, c0B, c1B;
        load_chunk<false>(values, rows, cols, keep_mask, k,     kEnd, r0, m, half,
                          inv_keep, aA, c0A, c1A);
        load_chunk<false>(values, rows, cols, keep_mask, k + 4, kEnd, r0, m, half,
                          inv_keep, aB, c0B, c1B);

        // Prefetch next iteration's gathered weight rows (global_prefetch_b8);
        // 16 lanes x 32B stride cover each 512B row. Clamped -> no branches.
        {
            int kp = k + 8 + 2 * half;
            int kq = kp + 4;
            kp = kp < kEnd ? kp : kEnd - 1;
            kq = kq < kEnd ? kq : kEnd - 1;
            const int cp = cols[kp];
            const int cq = cols[kq];
            __builtin_prefetch(weights + (size_t)cp * OUT_DIM + m * 8, 0, 3);
            __builtin_prefetch(weights + (size_t)cq * OUT_DIM + m * 8, 0, 3);
        }

        mma_chunk(weights, m, aA, c0A, c1A, acc);
        mma_chunk(weights, m, aB, c0B, c1B, acc);
    }

    // At most one remaining full K=4 chunk (still unguarded).
    if (k < kFull) {
        v2f a; int c0, c1;
        load_chunk<false>(values, rows, cols, keep_mask, k, kEnd, r0, m, half,
                          inv_keep, a, c0, c1);
        mma_chunk(weights, m, a, c0, c1, acc);
        k += 4;
    }

    // Tail chunk (<4 nnz): clamped indices + selects, no divergent branches.
    if (k < kEnd) {
        v2f a; int c0, c1;
        load_chunk<true>(values, rows, cols, keep_mask, k, kEnd, r0, m, half,
                         inv_keep, a, c0, c1);
        mma_chunk(weights, m, a, c0, c1, acc);
    }

    // Epilogue: bias add + store per the 16x16 f32 C layout
    // (VGPR j holds M=j for lanes 0-15, M=j+8 for lanes 16-31, N=lane&15).
#pragma unroll
    for (int nt = 0; nt < 8; ++nt) {
        const float bv = bias[nt * 16 + m];
#pragma unroll
        for (int j = 0; j < 8; ++j) {
            const int row = r0 + j + 8 * half;
            if (row < n_nodes) {
                out[(size_t)row * OUT_DIM + nt * 16 + m] = acc[nt][j] + bv;
            }
        }
    }
}

extern "C" void kernel_launch(void* const* d_in, const int* in_sizes, int n_in,
                              void* d_out, int out_size, void* d_ws, size_t ws_size,
                              hipStream_t stream) {
    // setup_inputs() order: values, rows, cols, keep_mask, weights, bias
    const float*         values    = (const float*)d_in[0];
    const int*           rows      = (const int*)d_in[1];
    const int*           cols      = (const int*)d_in[2];
    const unsigned char* keep_mask = (const unsigned char*)d_in[3]; // jax bool = 1 byte
    const float*         weights   = (const float*)d_in[4];
    const float*         bias      = (const float*)d_in[5];
    float*               out       = (float*)d_out;

    const int nnz     = in_sizes[0];
    const int n_nodes = out_size / OUT_DIM;
    const int ntiles  = (n_nodes + TILE_ROWS - 1) / TILE_ROWS;

    spmm_wmma_f32_kernel<<<ntiles, 32, 0, stream>>>(
        values, rows, cols, keep_mask, weights, bias, out, nnz, n_nodes);
}